// DEA1D_4904852652792
// MI455X (gfx1250) — compile-verified
//
#include <hip/hip_runtime.h>
#include <math.h>

// Problem constants
#define NB     32
#define NC     64
#define NE     4093          // encoder positions
#define DKW    16            // dictionary width
#define STR    4
#define NITER  20
#define EPAD   4104          // padded e-stride (4 left pad, tail pad)
#define PADL   4
#define GK     448           // Gram K dim = 7 deltas * 64 channels
#define GSTR   68            // LDS row stride for a 64x64 G slice (bank-conflict free)
#define NSTATE ((long)NC * EPAD * NB)     // floats per padded state buffer
#define XN_ELEMS (8382464L)  // 32*64*4093
#define Z_ELEMS  (524288L)   // 32*16384

typedef float  f32x2 __attribute__((ext_vector_type(2)));
typedef float  f32x8 __attribute__((ext_vector_type(8)));

// ---------------- zero workspace ----------------
__global__ void dea_zero(float4* __restrict__ p, long n4) {
    long i = (long)blockIdx.x * blockDim.x + threadIdx.x;
    long stride = (long)gridDim.x * blockDim.x;
    float4 z; z.x = 0.f; z.y = 0.f; z.z = 0.f; z.w = 0.f;
    for (; i < n4; i += stride) p[i] = z;
}

// ---------------- Gram tensor: G[c][ (d+3)*64 + c' ] = sum_k H[c,k]*H[c',k-4d] ----
__global__ void dea_gram(const float* __restrict__ H, float* __restrict__ G) {
    int t = blockIdx.x * 256 + threadIdx.x;       // 64*448 = 28672 threads
    if (t >= NC * GK) return;
    int c  = t / GK;
    int kk = t - c * GK;
    int d  = (kk >> 6) - 3;                       // delta in [-3,3]
    int cp = kk & 63;
    float s = 0.f;
    #pragma unroll
    for (int k = 0; k < DKW; ++k) {
        int k2 = k - 4 * d;
        if (k2 >= 0 && k2 < DKW) s += H[c * DKW + k] * H[cp * DKW + k2];
    }
    G[t] = s;
}

// ---------------- Hx = conv(x,H) (iteration-invariant) ----------------
__global__ void dea_conv_fwd(const float* __restrict__ x, const float* __restrict__ H,
                             float* __restrict__ Hx) {
    __shared__ float Hs[NC * DKW];
    for (int i = threadIdx.x; i < NC * DKW; i += 256) Hs[i] = H[i];
    __syncthreads();
    long t = (long)blockIdx.x * 256 + threadIdx.x;    // 32*64*4093
    if (t >= XN_ELEMS) return;
    int  e = (int)(t % NE);
    long q = t / NE;
    int  c = (int)(q & 63);
    int  b = (int)(q >> 6);
    const float* xp = x + (long)b * 16384 + 4 * e;
    const float* hp = Hs + c * DKW;
    float s = 0.f;
    #pragma unroll
    for (int k = 0; k < DKW; ++k) s += xp[k] * hp[k];
    Hx[((long)b * NC + c) * EPAD + PADL + e] = s;
}

// ---------------- One FISTA iteration (WMMA f32 16x16x4 over K=448) ----------
// Block: 256 thr = 8 waves; block covers 32 e-positions x 64 channels for one batch.
// Wave w: channel tile (w&3)*16, e tile blockIdx.x*2 + (w>>2).
__global__ __launch_bounds__(256)
void dea_fista_iter(const float* __restrict__ G,  const float* __restrict__ Hx,
                    const float* __restrict__ Yc, const float* __restrict__ Xc,
                    float* __restrict__ Yn, float* __restrict__ Xn,
                    float invL, float thr, float mom) {
    __shared__ float Gs[NC * GSTR];                 // one 64x64 delta-slice of G

    const int tid  = threadIdx.x;
    const int wave = tid >> 5;
    const int lane = tid & 31;
    const int hh   = lane >> 4;                     // half: 0 or 1
    const int ln   = lane & 15;
    const int c0   = (wave & 3) << 4;
    const int e0   = ((blockIdx.x << 1) + (wave >> 2)) << 4;
    const int b    = blockIdx.y;

    const float* Yb = Yc + (long)b * NC * EPAD;

    f32x8 acc = {0.f, 0.f, 0.f, 0.f, 0.f, 0.f, 0.f, 0.f};

    for (int dly = 0; dly < 7; ++dly) {             // delta+3
        __syncthreads();
        // stage 64x64 slice of G into LDS (row stride 68 -> conflict-free)
        for (int i = tid; i < NC * NC; i += 256) {
            int r  = i >> 6;
            int cp = i & 63;
            Gs[r * GSTR + cp] = G[r * GK + (dly << 6) + cp];
        }
        __syncthreads();

        // B rows live at e = e0 + n + (dly-3), padded so never OOB, pads are zero
        const float* yrow = Yb + (long)(2 * hh) * EPAD + (PADL - 3) + e0 + dly + ln;
        const float* arow = Gs + (c0 + ln) * GSTR + 2 * hh;

        #pragma unroll
        for (int kc = 0; kc < 16; ++kc) {           // c' in chunks of 4
            const int cp0 = kc << 2;
            f32x2 a, bf;
            a.x = arow[cp0];                        // A: K=k0+2*hh
            a.y = arow[cp0 + 1];                    // A: K=k0+2*hh+1
            const float* yp = yrow + (long)cp0 * EPAD;
            bf.x = yp[0];                           // B row K=k0+2*hh
            bf.y = yp[EPAD];                        // B row K=k0+2*hh+1
            acc = __builtin_amdgcn_wmma_f32_16x16x4_f32(
                      false, a, false, bf, (short)0, acc, false, false);
        }
    }

    // Epilogue: x_new = shrink(yk + (Hx - Gram)/L), momentum update
    const long bo = (long)b * NC * EPAD;
    const int  e  = e0 + ln;
    const bool valid = (e < NE);
    #pragma unroll
    for (int i = 0; i < 8; ++i) {
        const int  c   = c0 + i + 8 * hh;           // C/D layout: M=i / M=8+i
        const long idx = bo + (long)c * EPAD + PADL + e;
        float g  = acc[i];
        float hx = Hx[idx];
        float yv = Yc[idx];
        float xo = Xc[idx];
        float v  = yv + (hx - g) * invL;
        float av = fabsf(v) - thr;
        av = av > 0.f ? av : 0.f;
        float xn = copysignf(av, v);
        float yn = xn + mom * (xn - xo);
        if (valid) {
            Xn[idx] = xn;
            Yn[idx] = yn;
        }
    }
}

// ---------------- z = conv_transpose(x_final, H) ----------------
__global__ void dea_conv_t(const float* __restrict__ Xf, const float* __restrict__ H,
                           float* __restrict__ z) {
    __shared__ float Hs[NC * DKW];
    for (int i = threadIdx.x; i < NC * DKW; i += 256) Hs[i] = H[i];
    __syncthreads();
    long t = (long)blockIdx.x * 256 + threadIdx.x;   // 32*16384
    if (t >= Z_ELEMS) return;
    int b = (int)(t >> 14);
    int d = (int)(t & 16383);
    int p = d & 3, m = d >> 2;
    const float* Xb = Xf + (long)b * NC * EPAD;
    float s = 0.f;
    for (int c = 0; c < NC; ++c) {
        const float* xc = Xb + (long)c * EPAD + PADL + m;
        const float* hp = Hs + c * DKW + p;
        #pragma unroll
        for (int j = 0; j < 4; ++j) s += xc[-j] * hp[4 * j];
    }
    z[t] = s;
}

// ---------------- copy x_new (strip padding) + lambda ----------------
__global__ void dea_copy_out(const float* __restrict__ Xf, float* __restrict__ out,
                             float lam) {
    long t = (long)blockIdx.x * 256 + threadIdx.x;
    if (t == 0) out[XN_ELEMS] = lam;
    if (t >= XN_ELEMS) return;
    int  e  = (int)(t % NE);
    long ce = t / NE;                                // b*64 + c
    out[t] = Xf[ce * EPAD + PADL + e];
}

extern "C" void kernel_launch(void* const* d_in, const int* in_sizes, int n_in,
                              void* d_out, int out_size, void* d_ws, size_t ws_size,
                              hipStream_t stream) {
    (void)in_sizes; (void)n_in; (void)out_size; (void)ws_size;
    const float* x = (const float*)d_in[0];
    const float* H = (const float*)d_in[1];
    float* out = (float*)d_out;

    float* ya = (float*)d_ws;
    float* yb = ya + NSTATE;
    float* xa = yb + NSTATE;
    float* xb = xa + NSTATE;
    float* hx = xb + NSTATE;
    float* gm = hx + NSTATE;                         // 64*448 floats

    // host-side deterministic scalars
    const double lamd = 0.1 * sqrt(2.0 * log((double)(NC) * (double)NE));
    const float  lam  = (float)lamd;
    const float  thr  = (float)(lamd / 10.0);
    const float  invL = 0.1f;

    // init
    dea_zero<<<4096, 256, 0, stream>>>((float4*)ya, (5L * NSTATE) / 4);
    dea_gram<<<(NC * GK + 255) / 256, 256, 0, stream>>>(H, gm);
    dea_conv_fwd<<<(int)((XN_ELEMS + 255) / 256), 256, 0, stream>>>(x, H, hx);

    // FISTA iterations (ping-pong)
    float* Yc = ya; float* Xc = xa; float* Yn = yb; float* Xn = xb;
    double tk = 1.0;
    for (int it = 0; it < NITER; ++it) {
        double tn  = (1.0 + sqrt(1.0 + 4.0 * tk * tk)) * 0.5;
        float  mom = (float)((tk - 1.0) / tn);
        dea_fista_iter<<<dim3(128, NB), 256, 0, stream>>>(gm, hx, Yc, Xc, Yn, Xn,
                                                          invL, thr, mom);
        tk = tn;
        float* tmp;
        tmp = Yc; Yc = Yn; Yn = tmp;
        tmp = Xc; Xc = Xn; Xn = tmp;
    }

    // outputs: z, x_new, lam
    dea_conv_t<<<(int)((Z_ELEMS + 255) / 256), 256, 0, stream>>>(Xc, H, out);
    dea_copy_out<<<(int)((XN_ELEMS + 255) / 256), 256, 0, stream>>>(Xc, out + Z_ELEMS, lam);
}